// MyModel_87522843560542
// MI455X (gfx1250) — compile-verified
//
#include <hip/hip_runtime.h>
#include <math.h>

typedef __attribute__((ext_vector_type(2))) float v2f;
typedef __attribute__((ext_vector_type(4))) float v4f;
typedef __attribute__((ext_vector_type(8))) float v8f;

#define D2R 0.017453292519943295f
#define R2D 57.29577951308232f

// One wave (32 lanes) processes 4 samples. The 4x4 chain b <- b*T_j is done as
// V_WMMA_F32_16X16X4_F32 with 4 samples packed block-diagonally.
__global__ __launch_bounds__(256) void fk_wmma_kernel(
    const float* __restrict__ theta, const float* __restrict__ dh,
    float* __restrict__ out, int n)
{
    // Per-wave LDS: 6 joints * (4 samples * 16 floats) = 384 floats for T,
    // plus 64 floats for the diagonal-block staging buffer G (stored [s][col][row]).
    __shared__ float smem[8 * 448];
    const int lane = threadIdx.x & 31;
    const int wave = threadIdx.x >> 5;
    float* T = &smem[wave * 448];   // T[jnt*64 + s*16 + row*4 + col]
    float* G = T + 384;             // G[s*16 + col*4 + row]

    const int base = (blockIdx.x * 8 + wave) * 4;   // first sample of this wave

    // ---- Build the 6 per-sample DH matrices into LDS -----------------------
    // lane -> (sample s = lane/8, joint i = lane%8; i=6,7 idle)
    {
        const int s = lane >> 3;
        const int i = lane & 7;
        if (i < 6) {
            int samp = base + s; if (samp >= n) samp = n - 1;
            const float a     = dh[i * 4 + 0];
            const float alr   = dh[i * 4 + 1] * D2R;
            const float d     = dh[i * 4 + 2];
            const float off   = dh[i * 4 + 3];
            const float th    = (off + theta[samp * 6 + i]) * D2R;
            float st, ct, sa, ca;
            sincosf(th,  &st, &ct);
            sincosf(alr, &sa, &ca);
            float* tp = &T[i * 64 + s * 16];
            v4f r0; r0.x = ct;      r0.y = -st;     r0.z = 0.0f; r0.w = a;
            v4f r1; r1.x = st * ca; r1.y = ct * ca; r1.z = -sa;  r1.w = -sa * d;
            v4f r2; r2.x = st * sa; r2.y = ct * sa; r2.z = ca;   r2.w = ca * d;
            v4f r3; r3.x = 0.0f;    r3.y = 0.0f;    r3.z = 0.0f; r3.w = 1.0f;
            *(v4f*)(tp +  0) = r0;
            *(v4f*)(tp +  4) = r1;
            *(v4f*)(tp +  8) = r2;
            *(v4f*)(tp + 12) = r3;
        }
    }
    asm volatile("s_wait_dscnt 0" ::: "memory");   // same-wave DS in-order; cover VGPR hazard

    // Operand index helpers (WMMA f32 16x16x4 layouts, wave32):
    //  A: lane l holds row m=l&15, K = 2h+{0,1}, h = l>>4     (2 VGPRs)
    //  B: VGPR v, half h holds row K = 2h+v, N = l&15          (2 VGPRs)
    //  D: VGPR r, half h holds (M = r + 8h, N = l&15)          (8 VGPRs)
    const int h  = lane >> 4;
    const int m  = lane & 15;
    const int sA = m >> 2;    // sample slot (A-row block / B-col block)
    const int iA = m & 3;     // row-in-block for A, col-in-block for B

    // Initial accumulator A = T0 (A[m][k] = T0[row iA][col k], k = 2h+v)
    v2f Acc;
    Acc.x = T[0 * 64 + sA * 16 + iA * 4 + (2 * h + 0)];
    Acc.y = T[0 * 64 + sA * 16 + iA * 4 + (2 * h + 1)];

    // ---- Chain: b <- b * T_j, j = 1..5 (5 WMMAs) ---------------------------
    const bool wr = (h == ((m >> 3) & 1));   // lanes 0-7 own s=0,1; lanes 24-31 own s=2,3
    const bool hi = (sA & 1);                // which accumulator half holds our rows
    #pragma unroll
    for (int j = 1; j < 6; ++j) {
        v2f Bv;
        Bv.x = T[j * 64 + sA * 16 + (2 * h + 0) * 4 + iA];   // B[K=2h  ][N]
        Bv.y = T[j * 64 + sA * 16 + (2 * h + 1) * 4 + iA];   // B[K=2h+1][N]
        v8f c = {};
        c = __builtin_amdgcn_wmma_f32_16x16x4_f32(
                false, Acc, false, Bv, (short)0, c, false, false);
        // Extract diagonal 4x4 blocks: D[4s+i, N=m] for the lane owning column N.
        // M = 4*sA + i  ->  accumulator component 4*(sA&1) + i.
        if (wr) {
            v4f dv;
            dv.x = hi ? c[4] : c[0];
            dv.y = hi ? c[5] : c[1];
            dv.z = hi ? c[6] : c[2];
            dv.w = hi ? c[7] : c[3];
            *(v4f*)&G[sA * 16 + iA * 4] = dv;   // G[s][col=iA][row 0..3]
        }
        asm volatile("s_wait_dscnt 0" ::: "memory");
        if (j < 5) {  // rebuild A from G: A[m][k] = b[row iA][col k] = G[s][k][iA]
            Acc.x = G[sA * 16 + (2 * h + 0) * 4 + iA];
            Acc.y = G[sA * 16 + (2 * h + 1) * 4 + iA];
        }
    }

    // ---- Epilogue: position + Euler angles, one lane per sample ------------
    if (lane < 4 && (base + lane) < n) {
        const int s = lane;
        #define GM(r, cc) G[s * 16 + (cc) * 4 + (r)]
        const float x = GM(0, 3), y = GM(1, 3), z = GM(2, 3);
        const float T00 = GM(0, 0), T01 = GM(0, 1), T02 = GM(0, 2);
        const float T10 = GM(1, 0), T11 = GM(1, 1), T12 = GM(1, 2);
        const float T22 = GM(2, 2);
        #undef GM
        float Aa, Bb, Cc;
        if (fabsf(T12) <= 1e-6f && fabsf(T22) <= 1e-6f) {
            Aa = R2D * atan2f(T10, T11);
            Bb = R2D * atan2f(T02, T22);
            Cc = 0.0f;
        } else {
            const float An = atan2f(-T01, T00);
            float sAn, cAn;
            sincosf(An, &sAn, &cAn);
            Aa = R2D * An;
            Bb = R2D * atan2f(T02, cAn * T00 - sAn * T01);
            Cc = R2D * atan2f(-T12, T22);
        }
        float* o = out + (size_t)(base + s) * 6;
        o[0] = x; o[1] = y; o[2] = z; o[3] = Aa; o[4] = Bb; o[5] = Cc;
    }
}

extern "C" void kernel_launch(void* const* d_in, const int* in_sizes, int n_in,
                              void* d_out, int out_size, void* d_ws, size_t ws_size,
                              hipStream_t stream) {
    const float* theta = (const float*)d_in[0];
    const float* dh    = (const float*)d_in[1];
    float* out = (float*)d_out;
    const int n = in_sizes[0] / 6;                 // number of samples (1048576)
    const int samplesPerBlock = 32;                // 8 waves * 4 samples
    const int blocks = (n + samplesPerBlock - 1) / samplesPerBlock;
    fk_wmma_kernel<<<blocks, 256, 0, stream>>>(theta, dh, out, n);
}